// GCNLayer_8727373545860
// MI455X (gfx1250) — compile-verified
//
#include <hip/hip_runtime.h>
#include <hip/hip_bf16.h>

typedef __attribute__((ext_vector_type(2))) float v2f;
typedef __attribute__((ext_vector_type(8))) float v8f;

#define F_DIM 128  // F_IN == F_OUT == 128

// ---------------------------------------------------------------------------
// Phase 1: support = X @ W  (fp32 WMMA, 16x16 tile per wave, K=128)
// Grid: one block per 16-row M tile; 8 waves/block, wave w owns N tile w*16.
// ---------------------------------------------------------------------------
__global__ void __launch_bounds__(256)
gcn_gemm_wmma(const float* __restrict__ X, const float* __restrict__ W,
              float* __restrict__ S, int n_nodes) {
    const int lane = threadIdx.x & 31;
    const int wave = threadIdx.x >> 5;          // 0..7 -> N tile
    const int m0   = blockIdx.x * 16;
    const int n0   = wave * 16;

    const int m  = lane & 15;                   // row / col within tile
    const int kh = lane >> 4;                   // K half-select (0 or 1)

    // Clamp (defensive; N_NODES=100000 is an exact multiple of 16)
    const int mrowA = (m0 + m < n_nodes) ? (m0 + m) : (n_nodes - 1);
    const float* __restrict__ xrow = X + (size_t)mrowA * F_DIM;

    v8f c = {0.f, 0.f, 0.f, 0.f, 0.f, 0.f, 0.f, 0.f};

#pragma unroll
    for (int k0 = 0; k0 < F_DIM; k0 += 4) {
        const int ka = k0 + 2 * kh;             // this lane's K pair base
        // A fragment: 16x4 f32. Lanes 0-15: K={k0,k0+1}; lanes 16-31: K={k0+2,k0+3}
        v2f a = *(const v2f*)(xrow + ka);        // 8B-aligned (ka even, row 128-aligned)
        // B fragment: 4x16 f32, N striped across lanes, K pair per lane half
        v2f b;
        b.x = W[(size_t)(ka + 0) * F_DIM + n0 + m];
        b.y = W[(size_t)(ka + 1) * F_DIM + n0 + m];
        // D = A*B + C   (8 args: neg_a, A, neg_b, B, c_mod, C, reuse_a, reuse_b)
        c = __builtin_amdgcn_wmma_f32_16x16x4_f32(false, a, false, b,
                                                  (short)0, c, false, false);
    }

    // C/D layout: VGPR i -> row M = i + 8*(lane>>4); col N = lane&15
    const int nc   = lane & 15;
    const int mrow = (lane >> 4) * 8;
#pragma unroll
    for (int i = 0; i < 8; ++i) {
        const int r = m0 + mrow + i;
        if (r < n_nodes)
            S[(size_t)r * F_DIM + n0 + nc] = c[i];
    }
}

// ---------------------------------------------------------------------------
// Phase 2: COO SpMM scatter: out[row] += val * support[col]
// One wave per edge: 32 lanes x float4 = 128 features.
// ---------------------------------------------------------------------------
__global__ void __launch_bounds__(256)
gcn_spmm(const float* __restrict__ S, const int* __restrict__ rows,
         const int* __restrict__ cols, const float* __restrict__ vals,
         float* out, int n_edges) {
    const int e = blockIdx.x * (blockDim.x >> 5) + (threadIdx.x >> 5);
    if (e >= n_edges) return;
    const int lane = threadIdx.x & 31;

    const int   r = rows[e];
    const int   cidx = cols[e];
    const float v = vals[e];

    const float4 s = *(const float4*)(S + (size_t)cidx * F_DIM + lane * 4);
    float* o = out + (size_t)r * F_DIM + lane * 4;

    // Native global_atomic_add_f32 path (no CAS loop)
    unsafeAtomicAdd(o + 0, v * s.x);
    unsafeAtomicAdd(o + 1, v * s.y);
    unsafeAtomicAdd(o + 2, v * s.z);
    unsafeAtomicAdd(o + 3, v * s.w);
}

// ---------------------------------------------------------------------------
// Phase 3: out = relu(out + bias)
// ---------------------------------------------------------------------------
__global__ void __launch_bounds__(256)
gcn_bias_relu(float* __restrict__ out, const float* __restrict__ bias, int n4) {
    const int i = blockIdx.x * blockDim.x + threadIdx.x;
    if (i >= n4) return;
    float4 v = ((const float4*)out)[i];
    const int b0 = (i * 4) & (F_DIM - 1);
    v.x = fmaxf(v.x + bias[b0 + 0], 0.f);
    v.y = fmaxf(v.y + bias[b0 + 1], 0.f);
    v.z = fmaxf(v.z + bias[b0 + 2], 0.f);
    v.w = fmaxf(v.w + bias[b0 + 3], 0.f);
    ((float4*)out)[i] = v;
}

// ---------------------------------------------------------------------------
extern "C" void kernel_launch(void* const* d_in, const int* in_sizes, int n_in,
                              void* d_out, int out_size, void* d_ws, size_t ws_size,
                              hipStream_t stream) {
    const float* X    = (const float*)d_in[0];   // [N, 128]
    const int*   rows = (const int*)  d_in[1];   // [E]
    const int*   cols = (const int*)  d_in[2];   // [E]
    const float* vals = (const float*)d_in[3];   // [E]
    const float* W    = (const float*)d_in[4];   // [128, 128]
    const float* bias = (const float*)d_in[5];   // [128]

    const int n_nodes = in_sizes[0] / F_DIM;
    const int n_edges = in_sizes[1];

    float* S   = (float*)d_ws;                   // support [N, 128] (51.2 MB)
    float* out = (float*)d_out;

    // Zero accumulation target (harness poisons d_out).
    hipMemsetAsync(d_out, 0, (size_t)out_size * sizeof(float), stream);

    // 1) Dense transform via WMMA
    const int mtiles = (n_nodes + 15) / 16;
    gcn_gemm_wmma<<<mtiles, 256, 0, stream>>>(X, W, S, n_nodes);

    // 2) Edge scatter (wave per edge)
    const int waves_per_block = 256 / 32;
    const int eblocks = (n_edges + waves_per_block - 1) / waves_per_block;
    gcn_spmm<<<eblocks, 256, 0, stream>>>(S, rows, cols, vals, out, n_edges);

    // 3) Bias + ReLU
    const int n4 = out_size / 4;
    gcn_bias_relu<<<(n4 + 255) / 256, 256, 0, stream>>>(out, bias, n4);
}